// cGraspvaeLoss_21131239096966
// MI455X (gfx1250) — compile-verified
//
#include <hip/hip_runtime.h>
#include <hip/hip_bf16.h>
#include <math.h>

// Problem sizes (match reference)
#define B 16
#define H 778
#define N 8192
#define E 2314
#define Z 16

#define HPAD 784        // 49 * 16, hand verts padded to tile multiple
#define HT 49           // hand tiles of 16
#define NCHUNK 16       // obj chunks per (call, batch)
#define OBJ_PER_WG 512  // 8 waves * 4 tiles * 16
#define OT_PER_WAVE 4

typedef __attribute__((ext_vector_type(2))) float v2f;
typedef __attribute__((ext_vector_type(8))) float v8f;

// min across the 16 lanes of each half-wave (xor masks < 16 stay in-half)
__device__ __forceinline__ float wave_min16(float v) {
#pragma unroll
  for (int off = 1; off < 16; off <<= 1) v = fminf(v, __shfl_xor(v, off, 32));
  return v;
}

__device__ __forceinline__ void block_reduce_add(float v, float* acc) {
  __shared__ float s[8];
#pragma unroll
  for (int off = 16; off >= 1; off >>= 1) v += __shfl_xor(v, off, 32);
  int lane = threadIdx.x & 31, wv = threadIdx.x >> 5;
  if (lane == 0) s[wv] = v;
  __syncthreads();
  if (threadIdx.x == 0) {
    float t = 0.0f;
    int nw = (blockDim.x + 31) >> 5;
    for (int i = 0; i < nw; ++i) t += s[i];
    atomicAdd(acc, t);
  }
}

// ---------------------------------------------------------------------------
// ws layout: [0..15] float accumulators; [16 ..) 2*B*H uint h2o min-bits
// ---------------------------------------------------------------------------
__global__ void init_ws(unsigned* __restrict__ ws_u) {
  int i = blockIdx.x * blockDim.x + threadIdx.x;
  const int TOT = 16 + 2 * B * H;
  if (i < 16) ws_u[i] = 0u;                 // float 0.0 accumulators
  else if (i < TOT) ws_u[i] = 0x7F800000u;  // +inf (bits preserve order for d2>=0)
}

// ---------------------------------------------------------------------------
// WMMA distance kernel: per (call, batch, obj-chunk).
//   D = A x B + C with A=(-2hx,-2hy,-2hz,1)[16x4], B=(ox,oy,oz,o2)[4x16],
//   C=h2[row]  ->  D[m,n] = ||h_m||^2 + ||o_n||^2 - 2 h_m.o_n = d2
// ---------------------------------------------------------------------------
__global__ __launch_bounds__(256)
void dist_kernel(const float* __restrict__ rhand_vs,       // [B,3,H]
                 const float* __restrict__ rhand_pred,     // [B,H,3]
                 const float* __restrict__ obj_vs,         // [B,3,N]
                 const float* __restrict__ rh_mesh,        // [B,H,3]
                 const float* __restrict__ rh_mesh_pred,   // [B,H,3]
                 float* __restrict__ out,                  // d_out
                 unsigned* __restrict__ h2o_bits) {        // [2,B,H]
  __shared__ float s_hx[HPAD], s_hy[HPAD], s_hz[HPAD], s_h2[HPAD];
  __shared__ float s_nx[HPAD], s_ny[HPAD], s_nz[HPAD];

  const int wg = blockIdx.x;
  const int chunk = wg % NCHUNK;
  const int b = (wg / NCHUNK) % B;
  const int call = wg / (NCHUNK * B);   // 0: ground truth, 1: pred
  const int tid = threadIdx.x;

  // Stage hand verts + normals + ||h||^2 into LDS (pad rows huge & harmless)
  for (int h = tid; h < HPAD; h += 256) {
    float x, y, z, nx, ny, nz;
    if (h < H) {
      if (call == 0) {
        x = rhand_vs[b * 3 * H + 0 * H + h];
        y = rhand_vs[b * 3 * H + 1 * H + h];
        z = rhand_vs[b * 3 * H + 2 * H + h];
        nx = rh_mesh[(b * H + h) * 3 + 0];
        ny = rh_mesh[(b * H + h) * 3 + 1];
        nz = rh_mesh[(b * H + h) * 3 + 2];
      } else {
        x = rhand_pred[(b * H + h) * 3 + 0];
        y = rhand_pred[(b * H + h) * 3 + 1];
        z = rhand_pred[(b * H + h) * 3 + 2];
        nx = rh_mesh_pred[(b * H + h) * 3 + 0];
        ny = rh_mesh_pred[(b * H + h) * 3 + 1];
        nz = rh_mesh_pred[(b * H + h) * 3 + 2];
      }
    } else {
      x = y = z = 1.0e6f;  // pad: d2 ~ 3e12, finite, never the argmin
      nx = ny = nz = 1.0f;
    }
    s_hx[h] = x; s_hy[h] = y; s_hz[h] = z;
    s_h2[h] = x * x + y * y + z * z;
    s_nx[h] = nx; s_ny[h] = ny; s_nz[h] = nz;
  }
  __syncthreads();

  const int wave = tid >> 5, lane = tid & 31;
  const int half = lane >> 4;   // 0 -> K={0,1}, rows 0-7 ; 1 -> K={2,3}, rows 8-15
  const int lc = lane & 15;     // obj column within tile / hand M within tile
  const int objbase = chunk * OBJ_PER_WG + wave * (OT_PER_WAVE * 16);
  const float* objb = obj_vs + (size_t)b * 3 * N;

  // B operands: K-striped per 16x16x4 layout (lanes 0-15: K0,K1; 16-31: K2,K3)
  v2f bop[OT_PER_WAVE];
#pragma unroll
  for (int t = 0; t < OT_PER_WAVE; ++t) {
    int n = objbase + t * 16 + lc;
    float ox = objb[0 * N + n], oy = objb[1 * N + n], oz = objb[2 * N + n];
    float o2 = ox * ox + oy * oy + oz * oz;
    bop[t].x = half ? oz : ox;
    bop[t].y = half ? o2 : oy;
  }

  float minv[OT_PER_WAVE];
  int mini[OT_PER_WAVE];
#pragma unroll
  for (int t = 0; t < OT_PER_WAVE; ++t) { minv[t] = 3.0e38f; mini[t] = 0; }

  for (int ht = 0; ht < HT; ++ht) {
    int m = ht * 16 + lc;
    v2f a;
    a.x = half ? (-2.0f * s_hz[m]) : (-2.0f * s_hx[m]);
    a.y = half ? 1.0f              : (-2.0f * s_hy[m]);
    const int rowbase = ht * 16 + half * 8;
    v8f c;
#pragma unroll
    for (int r = 0; r < 8; ++r) c[r] = s_h2[rowbase + r];

    float hmin[8];
#pragma unroll
    for (int r = 0; r < 8; ++r) hmin[r] = 3.0e38f;

#pragma unroll
    for (int t = 0; t < OT_PER_WAVE; ++t) {
      v8f d = __builtin_amdgcn_wmma_f32_16x16x4_f32(
          /*neg_a=*/false, a, /*neg_b=*/false, bop[t],
          /*c_mod=*/(short)0, c, /*reuse_a=*/false, /*reuse_b=*/false);
#pragma unroll
      for (int r = 0; r < 8; ++r) {
        float v = fmaxf(d[r], 0.0f);            // jnp.maximum(d2, 0)
        hmin[r] = fminf(hmin[r], v);            // per-hand min over obj
        if (v < minv[t]) { minv[t] = v; mini[t] = rowbase + r; }  // per-obj argmin
      }
    }

    // per-hand-row min across this wave's 64 obj points -> global atomic min
#pragma unroll
    for (int r = 0; r < 8; ++r) {
      float hv = wave_min16(hmin[r]);
      if (lc == r) {
        int row = rowbase + r;
        if (row < H)
          atomicMin(&h2o_bits[(call * B + b) * H + row], __float_as_uint(hv));
      }
    }
  }

  // Finish per-obj argmin: combine row halves, tie -> lower index (JAX argmin)
  // Output order: [6 scalars][o2h_s_p (call 1)][o2h_s (call 0)]
  float* outp = out + 6 + (size_t)(1 - call) * (B * N) + (size_t)b * N;
#pragma unroll
  for (int t = 0; t < OT_PER_WAVE; ++t) {
    float ov = minv[t]; int oi = mini[t];
    float ov2 = __shfl_xor(ov, 16, 32);
    int   oi2 = __shfl_xor(oi, 16, 32);
    if (ov2 < ov || (ov2 == ov && oi2 < oi)) { ov = ov2; oi = oi2; }
    if (lane < 16) {
      int n = objbase + t * 16 + lane;
      float ox = objb[0 * N + n], oy = objb[1 * N + n], oz = objb[2 * N + n];
      float dx = ox - s_hx[oi], dy = oy - s_hy[oi], dz = oz - s_hz[oi];
      float dist = sqrtf(dx * dx + dy * dy + dz * dz + 1e-12f);
      float s = s_nx[oi] * dx + s_ny[oi] * dy + s_nz[oi] * dz;
      float sgn = (s > 0.0f) ? 1.0f : ((s < 0.0f) ? -1.0f : 0.0f);
      outp[n] = dist * sgn;
    }
  }
}

// ---------------------------------------------------------------------------
// Loss reduction kernels
// ---------------------------------------------------------------------------
__global__ void h2o_loss_kernel(const unsigned* __restrict__ bits,
                                const float* __restrict__ vw,
                                float* __restrict__ acc) {
  int i = blockIdx.x * blockDim.x + threadIdx.x;
  float v = 0.0f;
  if (i < B * H) {
    int h = i % H;
    float h2o  = sqrtf(__uint_as_float(bits[i]) + 1e-12f);          // call 0
    float h2op = sqrtf(__uint_as_float(bits[B * H + i]) + 1e-12f);  // call 1
    float w2 = powf(vw[h], 0.4f);  // v_weights ** (1/2.5)
    v = fabsf(h2op - h2o) * w2;
  }
  block_reduce_add(v, acc + 0);
}

__global__ void o2h_loss_kernel(const float* __restrict__ out,
                                const float* __restrict__ region,
                                float* __restrict__ acc) {
  int i = blockIdx.x * blockDim.x + threadIdx.x;
  float v = 0.0f;
  if (i < B * N) {
    float sp = out[6 + i];           // o2h_s_p
    float s  = out[6 + B * N + i];   // o2h_s
    float w = ((s < 0.01f) && (s > -0.005f)) ? 1.0f : 0.1f;  // W_CONTACT overwrite
    if (sp < 0.0f) w = 1.5f;                                  // W_PENET
    if (region[i] > 0.0f) w = 2.0f;                           // W_REGION
    v = fabsf(sp - s) * w;
  }
  block_reduce_add(v, acc + 1);
}

__global__ void kl_kernel(const float* __restrict__ mu,
                          const float* __restrict__ st,
                          float* __restrict__ acc) {
  int i = threadIdx.x;
  float v = 0.0f;
  if (i < B * Z) {
    float x = st[i];
    float sig = (x > 20.0f) ? x : log1pf(expf(x));  // softplus
    float m = mu[i];
    v = -logf(sig) + 0.5f * (sig * sig + m * m - 1.0f);
  }
  block_reduce_add(v, acc + 2);
}

__global__ void mesh_kernel(const float* __restrict__ rhand_vs,
                            const float* __restrict__ pred,
                            const float* __restrict__ vw,
                            float* __restrict__ acc) {
  int i = blockIdx.x * blockDim.x + threadIdx.x;
  float v = 0.0f;
  if (i < B * H * 3) {
    int b = i / (H * 3);
    int r = i % (H * 3);
    int h = r / 3, k = r % 3;
    float g = rhand_vs[b * 3 * H + k * H + h];  // transposed ground truth
    float p = pred[i];
    v = fabsf(g - p) * vw[h];
  }
  block_reduce_add(v, acc + 3);
}

__global__ void edge_kernel(const float* __restrict__ rhand_vs,
                            const float* __restrict__ pred,
                            const int* __restrict__ vpe,
                            float* __restrict__ acc) {
  int i = blockIdx.x * blockDim.x + threadIdx.x;
  float v = 0.0f;
  if (i < B * E * 3) {
    int b = i / (E * 3);
    int r = i % (E * 3);
    int e = r / 3, k = r % 3;
    int i0 = vpe[e * 2 + 0], i1 = vpe[e * 2 + 1];
    float ep = pred[(b * H + i0) * 3 + k] - pred[(b * H + i1) * 3 + k];
    float eg = rhand_vs[b * 3 * H + k * H + i0] - rhand_vs[b * 3 * H + k * H + i1];
    v = fabsf(ep - eg);
  }
  block_reduce_add(v, acc + 4);
}

__global__ void combine_kernel(const float* __restrict__ acc, float* __restrict__ out) {
  if (threadIdx.x == 0 && blockIdx.x == 0) {
    float dh   = 35.0f  * acc[0] / (float)(B * H);
    float dofl = 30.0f  * acc[1] / (float)(B * N);
    float kl   = 0.005f * acc[2] / (float)B;          // mean over batch of sum over Z
    float mesh = 30.0f  * acc[3] / (float)(B * H * 3);
    float edge = 30.0f  * acc[4] / (float)(B * E * 3);
    out[0] = kl + edge + mesh + dh + dofl;
    out[1] = kl;
    out[2] = edge;
    out[3] = mesh;
    out[4] = dh;
    out[5] = dofl;
  }
}

// ---------------------------------------------------------------------------
extern "C" void kernel_launch(void* const* d_in, const int* in_sizes, int n_in,
                              void* d_out, int out_size, void* d_ws, size_t ws_size,
                              hipStream_t stream) {
  (void)in_sizes; (void)n_in; (void)out_size; (void)ws_size;
  const float* rhand_vs     = (const float*)d_in[0];  // [B,3,H]
  const float* rhand_pred   = (const float*)d_in[1];  // [B,H,3]
  const float* obj_vs       = (const float*)d_in[2];  // [B,3,N]
  const float* rh_mesh      = (const float*)d_in[3];  // [B,H,3]
  const float* rh_mesh_pred = (const float*)d_in[4];  // [B,H,3]
  const float* region       = (const float*)d_in[5];  // [B,1,N]
  const float* p_mean       = (const float*)d_in[6];  // [B,Z]
  const float* p_std        = (const float*)d_in[7];  // [B,Z]
  const float* v_weights    = (const float*)d_in[8];  // [H]
  const int*   vpe          = (const int*)d_in[9];    // [E,2]

  float* out = (float*)d_out;
  float* acc = (float*)d_ws;                 // 5 partial-sum accumulators
  unsigned* bits = (unsigned*)d_ws + 16;     // [2,B,H] h2o min bits

  const int initN = 16 + 2 * B * H;
  init_ws<<<(initN + 255) / 256, 256, 0, stream>>>((unsigned*)d_ws);

  dist_kernel<<<2 * B * NCHUNK, 256, 0, stream>>>(
      rhand_vs, rhand_pred, obj_vs, rh_mesh, rh_mesh_pred, out, bits);

  h2o_loss_kernel<<<(B * H + 255) / 256, 256, 0, stream>>>(bits, v_weights, acc);
  o2h_loss_kernel<<<(B * N + 255) / 256, 256, 0, stream>>>(out, region, acc);
  kl_kernel<<<1, 256, 0, stream>>>(p_mean, p_std, acc);
  mesh_kernel<<<(B * H * 3 + 255) / 256, 256, 0, stream>>>(rhand_vs, rhand_pred, v_weights, acc);
  edge_kernel<<<(B * E * 3 + 255) / 256, 256, 0, stream>>>(rhand_vs, rhand_pred, vpe, acc);
  combine_kernel<<<1, 1, 0, stream>>>(acc, out);
}